// GotenNetEmbedding_25177098289382
// MI455X (gfx1250) — compile-verified
//
#include <hip/hip_runtime.h>

typedef float v2f __attribute__((ext_vector_type(2)));
typedef float v8f __attribute__((ext_vector_type(8)));

#define WMMA_F32(a, b, c) \
    __builtin_amdgcn_wmma_f32_16x16x4_f32(false, (a), false, (b), (short)0, (c), false, false)

constexpr int NN = 50000;   // nodes
constexpr int NE = 800000;  // edges
constexpr int D  = 128;     // node dim
constexpr int ED = 32;      // edge radial dim
constexpr int APAD = 132;   // LDS row stride (128 + 4) -> conflict-free column reads
constexpr int EPAD = 36;    // LDS row stride for 32-wide edge tile

// ---------------------------------------------------------------------------
// Zero m_node accumulator (ws is poisoned by harness; must re-zero every call)
// ---------------------------------------------------------------------------
__global__ void zero_kernel(float* __restrict__ p, int n) {
    int i = blockIdx.x * blockDim.x + threadIdx.x;
    if (i < n) p[i] = 0.0f;
}

// ---------------------------------------------------------------------------
// proj_node = node_attr @ W_node ; proj_center = node_attr @ W_center_node
// One block = one 16-row node tile, staged in LDS. 16 waves: w<8 -> W_node
// column tile w, w>=8 -> W_center column tile (w-8). K-loop of 32 fp32 WMMAs.
// ---------------------------------------------------------------------------
__global__ __launch_bounds__(512)
void node_proj_kernel(const float* __restrict__ node_attr,
                      const float* __restrict__ Wn,
                      const float* __restrict__ Wc,
                      float* __restrict__ proj_node,
                      float* __restrict__ proj_center) {
    __shared__ float At[16 * APAD];
    const int tile = blockIdx.x;
    const int tid  = threadIdx.x;

    for (int i = tid; i < 16 * D; i += 512) {
        int r = i >> 7, c = i & 127;
        At[r * APAD + c] = node_attr[(size_t)(tile * 16 + r) * D + c];
    }
    __syncthreads();

    const int w    = tid >> 5;
    const int lane = tid & 31;
    const int half = lane >> 4;   // 0: lanes 0-15, 1: lanes 16-31
    const int l    = lane & 15;
    const float* __restrict__ W   = (w < 8) ? Wn : Wc;
    float* __restrict__       out = (w < 8) ? proj_node : proj_center;
    const int tn = w & 7;         // output column tile

    v8f acc = {};
    for (int k0 = 0; k0 < D; k0 += 4) {
        const int k = k0 + 2 * half;
        v2f a, b;
        a.x = At[l * APAD + k];
        a.y = At[l * APAD + k + 1];
        b.x = W[(size_t)k * D + tn * 16 + l];
        b.y = W[(size_t)(k + 1) * D + tn * 16 + l];
        acc = WMMA_F32(a, b, acc);
    }
#pragma unroll
    for (int j = 0; j < 8; ++j) {
        const int row = tile * 16 + j + 8 * half;
        out[(size_t)row * D + tn * 16 + l] = acc[j];
    }
}

// ---------------------------------------------------------------------------
// Fused edge kernel: proj_radial tile = edge_attr_tile @ W_edge (WMMA, kept in
// regs), then gather proj_node[neigh], multiply, atomic-scatter into
// m_node[center]. One block = 16 edges, 8 waves = 8 column tiles of 16.
// ---------------------------------------------------------------------------
__global__ __launch_bounds__(256)
void edge_msg_kernel(const float* __restrict__ edge_attr,
                     const long long* __restrict__ eidx,
                     const float* __restrict__ We,
                     const float* __restrict__ proj_node,
                     float* __restrict__ m_node) {
    __shared__ float Et[16 * EPAD];
    __shared__ int centers[16];
    __shared__ int neighs[16];
    const int tile = blockIdx.x;
    const int tid  = threadIdx.x;

    for (int i = tid; i < 16 * ED; i += 256) {
        int r = i >> 5, c = i & 31;
        Et[r * EPAD + c] = edge_attr[(size_t)(tile * 16 + r) * ED + c];
    }
    if (tid < 16) {
        centers[tid] = (int)eidx[(size_t)tile * 16 + tid];          // edge_index[0]
        neighs[tid]  = (int)eidx[(size_t)NE + tile * 16 + tid];     // edge_index[1]
    }
    __syncthreads();

    const int w    = tid >> 5;
    const int lane = tid & 31;
    const int half = lane >> 4;
    const int l    = lane & 15;

    v8f acc = {};
    for (int k0 = 0; k0 < ED; k0 += 4) {
        const int k = k0 + 2 * half;
        v2f a, b;
        a.x = Et[l * EPAD + k];
        a.y = Et[l * EPAD + k + 1];
        b.x = We[(size_t)k * D + w * 16 + l];
        b.y = We[(size_t)(k + 1) * D + w * 16 + l];
        acc = WMMA_F32(a, b, acc);
    }

    const int col = w * 16 + l;
#pragma unroll
    for (int j = 0; j < 8; ++j) {
        const int row = j + 8 * half;          // edge within tile
        const float v = acc[j] * proj_node[(size_t)neighs[row] * D + col];
        unsafeAtomicAdd(&m_node[(size_t)centers[row] * D + col], v);
    }
}

// ---------------------------------------------------------------------------
// h = LN( [proj_center, m_node] @ W_concat ) : two K=128 WMMA loops into one
// accumulator, result staged to LDS, then wave-shuffle LayerNorm (16 lanes per
// row, 8 columns per lane) and store.
// ---------------------------------------------------------------------------
__global__ __launch_bounds__(256)
void out_kernel(const float* __restrict__ proj_center,
                const float* __restrict__ m_node,
                const float* __restrict__ Wcat,
                const float* __restrict__ lnw,
                const float* __restrict__ lnb,
                float* __restrict__ out) {
    __shared__ float Ct[16 * APAD];
    __shared__ float Mt[16 * APAD];
    __shared__ float Ht[16 * APAD];
    const int tile = blockIdx.x;
    const int tid  = threadIdx.x;

    for (int i = tid; i < 16 * D; i += 256) {
        int r = i >> 7, c = i & 127;
        Ct[r * APAD + c] = proj_center[(size_t)(tile * 16 + r) * D + c];
        Mt[r * APAD + c] = m_node[(size_t)(tile * 16 + r) * D + c];
    }
    __syncthreads();

    const int w    = tid >> 5;
    const int lane = tid & 31;
    const int half = lane >> 4;
    const int l    = lane & 15;

    v8f acc = {};
    for (int k0 = 0; k0 < D; k0 += 4) {          // top half: proj_center @ Wcat[0:128]
        const int k = k0 + 2 * half;
        v2f a, b;
        a.x = Ct[l * APAD + k];
        a.y = Ct[l * APAD + k + 1];
        b.x = Wcat[(size_t)k * D + w * 16 + l];
        b.y = Wcat[(size_t)(k + 1) * D + w * 16 + l];
        acc = WMMA_F32(a, b, acc);
    }
    for (int k0 = 0; k0 < D; k0 += 4) {          // bottom half: m_node @ Wcat[128:256]
        const int k = k0 + 2 * half;
        v2f a, b;
        a.x = Mt[l * APAD + k];
        a.y = Mt[l * APAD + k + 1];
        b.x = Wcat[(size_t)(D + k) * D + w * 16 + l];
        b.y = Wcat[(size_t)(D + k + 1) * D + w * 16 + l];
        acc = WMMA_F32(a, b, acc);
    }
#pragma unroll
    for (int j = 0; j < 8; ++j)
        Ht[(j + 8 * half) * APAD + w * 16 + l] = acc[j];
    __syncthreads();

    // LayerNorm: 256 threads = 16 rows x 16 sub-lanes; each lane owns 8 cols.
    const int row = tid >> 4;
    const int sub = tid & 15;
    float x[8];
    float s = 0.0f;
#pragma unroll
    for (int i = 0; i < 8; ++i) { x[i] = Ht[row * APAD + sub + 16 * i]; s += x[i]; }
#pragma unroll
    for (int m = 1; m < 16; m <<= 1) s += __shfl_xor(s, m, 32);
    const float mu = s * (1.0f / 128.0f);
    float vs = 0.0f;
#pragma unroll
    for (int i = 0; i < 8; ++i) { const float d = x[i] - mu; vs += d * d; }
#pragma unroll
    for (int m = 1; m < 16; m <<= 1) vs += __shfl_xor(vs, m, 32);
    const float inv = rsqrtf(vs * (1.0f / 128.0f) + 1e-5f);
#pragma unroll
    for (int i = 0; i < 8; ++i) {
        const int c = sub + 16 * i;
        out[(size_t)(tile * 16 + row) * D + c] = (x[i] - mu) * inv * lnw[c] + lnb[c];
    }
}

// ---------------------------------------------------------------------------
extern "C" void kernel_launch(void* const* d_in, const int* in_sizes, int n_in,
                              void* d_out, int out_size, void* d_ws, size_t ws_size,
                              hipStream_t stream) {
    const float*     node_attr = (const float*)d_in[0];
    const float*     edge_attr = (const float*)d_in[1];
    const long long* eidx      = (const long long*)d_in[2];
    const float*     W_node    = (const float*)d_in[3];
    const float*     W_center  = (const float*)d_in[4];
    const float*     W_cat     = (const float*)d_in[5];
    const float*     W_edge    = (const float*)d_in[6];
    const float*     lnw       = (const float*)d_in[7];
    const float*     lnb       = (const float*)d_in[8];
    float*           out       = (float*)d_out;

    float* ws          = (float*)d_ws;
    float* proj_node   = ws;                         // [NN, D]
    float* proj_center = ws + (size_t)NN * D;        // [NN, D]
    float* m_node      = ws + (size_t)2 * NN * D;    // [NN, D]

    zero_kernel<<<(NN * D + 255) / 256, 256, 0, stream>>>(m_node, NN * D);
    node_proj_kernel<<<NN / 16, 512, 0, stream>>>(node_attr, W_node, W_center,
                                                  proj_node, proj_center);
    edge_msg_kernel<<<NE / 16, 256, 0, stream>>>(edge_attr, eidx, W_edge,
                                                 proj_node, m_node);
    out_kernel<<<NN / 16, 256, 0, stream>>>(proj_center, m_node, W_cat,
                                            lnw, lnb, out);
}